// MoE_11398843204187
// MI455X (gfx1250) — compile-verified
//
#include <hip/hip_runtime.h>
#include <hip/hip_bf16.h>

#define TOKENS      4096
#define DMODEL      1024
#define N_EXPERTS   32
#define EXPERT_SIZE 512
#define N_HEADS     4
#define CAP         TOKENS          // worst-case tokens per expert
#define MTILE       32              // tokens per expert-GEMM block

typedef __attribute__((ext_vector_type(2))) float v2f;
typedef __attribute__((ext_vector_type(8))) float v8f;

// ---------------------------------------------------------------------------
// Kernel 0: zero output + small workspace regions
// ---------------------------------------------------------------------------
__global__ void moe_zero_kernel(float* __restrict__ out, int n,
                                float* __restrict__ S, int* __restrict__ cnt) {
    int i = blockIdx.x * blockDim.x + threadIdx.x;
    if (i < n) out[i] = 0.0f;
    if (i < N_EXPERTS) { S[i] = 0.0f; cnt[i] = 0; }
}

// ---------------------------------------------------------------------------
// Kernel 1: routing.  One wave32 per token, lane == expert.
//  - logits[t][e] = dot(x[t], expert_sel[e])
//  - accumulate exp(log_softmax) into S[e]  (entropy regularizer)
//  - sigmoid + top-4, append (token, gate) to per-expert dispatch lists
// ---------------------------------------------------------------------------
__global__ __launch_bounds__(256)
void moe_routing_kernel(const float* __restrict__ x,
                        const float* __restrict__ es,   // [E, d]
                        float* __restrict__ S,          // [E] accumulator
                        int*   __restrict__ cnt,        // [E]
                        int*   __restrict__ toklist,    // [E, CAP]
                        float* __restrict__ wlist) {    // [E, CAP]
    __shared__ float esT[DMODEL * N_EXPERTS];  // transposed: esT[k*32 + e], 128KB
    __shared__ float sblk[N_EXPERTS];

    const int tid  = threadIdx.x;
    const int wave = tid >> 5;
    const int lane = tid & 31;

    // stage expert_sel transposed into LDS (coalesced global reads)
    for (int i = tid; i < N_EXPERTS * DMODEL; i += 256) {
        int e = i >> 10, k = i & (DMODEL - 1);
        esT[k * N_EXPERTS + e] = es[i];
    }
    if (tid < N_EXPERTS) sblk[tid] = 0.0f;
    __syncthreads();

    const int t = blockIdx.x * 8 + wave;    // 512 blocks * 8 waves = 4096 tokens
    const float* xr = x + (size_t)t * DMODEL;

    // logits: lane e accumulates dot(x[t], expert_sel[e])
    float acc = 0.0f;
    #pragma unroll 8
    for (int k = 0; k < DMODEL; ++k) {
        float xk = xr[k];                       // wave-uniform (scalar load)
        acc = fmaf(xk, esT[k * N_EXPERTS + lane], acc);  // conflict-free LDS
    }

    // logsumexp over the 32 experts
    float m = acc;
    for (int off = 16; off > 0; off >>= 1)
        m = fmaxf(m, __shfl_xor(m, off, 32));
    float ex = __expf(acc - m);
    float ssum = ex;
    for (int off = 16; off > 0; off >>= 1)
        ssum += __shfl_xor(ssum, off, 32);
    float lse = m + __logf(ssum);
    float p = __expf(acc - lse);                // exp(log_softmax)

    // accumulate softmax mass per expert (block-local, then one atomic/expert)
    atomicAdd(&sblk[lane], p);

    // sigmoid gate + iterative top-4 argmax across lanes
    float sig = 1.0f / (1.0f + __expf(-acc));
    float v = sig;
    #pragma unroll
    for (int h = 0; h < N_HEADS; ++h) {
        float bv = v; int bi = lane;
        for (int off = 16; off > 0; off >>= 1) {
            float ov = __shfl_xor(bv, off, 32);
            int   oi = __shfl_xor(bi, off, 32);
            if (ov > bv || (ov == bv && oi < bi)) { bv = ov; bi = oi; }
        }
        if (lane == 0) {
            int pos = atomicAdd(&cnt[bi], 1);
            toklist[bi * CAP + pos] = t;
            wlist[bi * CAP + pos]   = bv;
        }
        if (lane == bi) v = -1e30f;             // knock out the winner
    }

    __syncthreads();
    if (wave == 0) atomicAdd(&S[lane], sblk[lane]);
}

// ---------------------------------------------------------------------------
// Kernel 2: finalize regularizer:  reg = sum_e exp(lm)*lm,  lm = log(S/T)
// ---------------------------------------------------------------------------
__global__ void moe_reg_finalize(const float* __restrict__ S,
                                 float* __restrict__ out_reg) {
    int lane = threadIdx.x;
    float s  = S[lane];
    float lm = __logf(s) - __logf((float)TOKENS);
    float c  = __expf(lm) * lm;
    for (int off = 16; off > 0; off >>= 1)
        c += __shfl_xor(c, off, 32);
    if (lane == 0) out_reg[0] = c;
}

// ---------------------------------------------------------------------------
// Kernel 3: grouped expert GEMM via V_WMMA_F32_16X16X4_F32.
// Block = (expert e, 32-token tile).  16 waves / block, 512 threads.
// Each wave keeps TWO 16x16 accumulators (M 0-15, 16-31) so every B fragment
// fetched from L2 feeds 32 token rows (2x arithmetic intensity vs 16-row tile).
//   GEMM1: scores[32 x 512] = relu(Xg[32 x 1024] @ keys[e]) * gate
//   GEMM2: out += scores @ values[e][512 x 1024]   (global f32 atomics)
// LDS: ~193 KB -- only possible on CDNA5's 320KB-per-workgroup LDS.
// ---------------------------------------------------------------------------
#define XPITCH 1026   // 1024 + 2 (bank-conflict padding)
#define SPITCH 516    // 512 + 4

__global__ __launch_bounds__(512)
void moe_expert_gemm(const float* __restrict__ x,
                     const float* __restrict__ keys,    // [E, d, h]
                     const float* __restrict__ values,  // [E, h, d]
                     const int*   __restrict__ cnt,
                     const int*   __restrict__ toklist,
                     const float* __restrict__ wlist,
                     float* __restrict__ out) {
    const int e  = blockIdx.x;
    const int m0 = blockIdx.y * MTILE;
    const int ncount = cnt[e];
    if (m0 >= ncount) return;
    const int rows = min(MTILE, ncount - m0);

    __shared__ float Xs[MTILE * XPITCH];   // gathered token rows (~128KB)
    __shared__ float Ss[MTILE * SPITCH];   // intermediate scores (~64.5KB)
    __shared__ float Ws[MTILE];
    __shared__ int   Ts[MTILE];

    const int tid  = threadIdx.x;
    const int wave = tid >> 5;
    const int lane = tid & 31;

    if (tid < MTILE) {
        int r = tid;
        if (r < rows) {
            Ts[r] = toklist[e * CAP + m0 + r];
            Ws[r] = wlist[e * CAP + m0 + r];
        } else {
            Ts[r] = -1;
            Ws[r] = 0.0f;
        }
    }
    __syncthreads();

    // gather X rows into LDS with b128 loads
    for (int i = tid; i < MTILE * (DMODEL / 4); i += 512) {
        int r = i >> 8;                 // DMODEL/4 == 256 per row
        int k = (i & 255) << 2;
        int t = Ts[r];
        float4 vv = make_float4(0.f, 0.f, 0.f, 0.f);
        if (t >= 0)
            vv = *(const float4*)(x + (size_t)t * DMODEL + k);
        Xs[r * XPITCH + k + 0] = vv.x;
        Xs[r * XPITCH + k + 1] = vv.y;
        Xs[r * XPITCH + k + 2] = vv.z;
        Xs[r * XPITCH + k + 3] = vv.w;
    }
    __syncthreads();

    const int mrow  = lane & 15;            // A-fragment row (within 16-row half)
    const int khalf = (lane >> 4) << 1;     // 0 (lanes 0-15) or 2 (lanes 16-31)
    const int ncol  = lane & 15;            // B/C-fragment column

    // ---------------- GEMM1: Xg @ keys[e], K = 1024 ----------------
    const float* kB = keys + (size_t)e * DMODEL * EXPERT_SIZE;
    for (int nt = wave; nt < EXPERT_SIZE / 16; nt += 16) {
        const int n0 = nt * 16;
        v8f c0 = {0.f, 0.f, 0.f, 0.f, 0.f, 0.f, 0.f, 0.f};
        v8f c1 = {0.f, 0.f, 0.f, 0.f, 0.f, 0.f, 0.f, 0.f};
        #pragma unroll 4
        for (int k0 = 0; k0 < DMODEL; k0 += 4) {
            v2f b, a0, a1;
            b.x  = kB[(size_t)(k0 + khalf + 0) * EXPERT_SIZE + n0 + ncol];
            b.y  = kB[(size_t)(k0 + khalf + 1) * EXPERT_SIZE + n0 + ncol];
            a0.x = Xs[mrow * XPITCH + k0 + khalf + 0];
            a0.y = Xs[mrow * XPITCH + k0 + khalf + 1];
            a1.x = Xs[(mrow + 16) * XPITCH + k0 + khalf + 0];
            a1.y = Xs[(mrow + 16) * XPITCH + k0 + khalf + 1];
            c0 = __builtin_amdgcn_wmma_f32_16x16x4_f32(
                     false, a0, false, b, (short)0, c0, false, false);
            c1 = __builtin_amdgcn_wmma_f32_16x16x4_f32(
                     false, a1, false, b, (short)0, c1, false, false);
        }
        // relu * gate, stash both 16-row halves into LDS scores tile
        #pragma unroll
        for (int r = 0; r < 8; ++r) {
            int M = r + ((lane >> 4) << 3);
            Ss[M * SPITCH + n0 + ncol]        = fmaxf(c0[r], 0.0f) * Ws[M];
            Ss[(M + 16) * SPITCH + n0 + ncol] = fmaxf(c1[r], 0.0f) * Ws[M + 16];
        }
    }
    __syncthreads();

    // ---------------- GEMM2: scores @ values[e], K = 512 ----------------
    const float* vB = values + (size_t)e * EXPERT_SIZE * DMODEL;
    for (int nt = wave; nt < DMODEL / 16; nt += 16) {
        const int n0 = nt * 16;
        v8f c0 = {0.f, 0.f, 0.f, 0.f, 0.f, 0.f, 0.f, 0.f};
        v8f c1 = {0.f, 0.f, 0.f, 0.f, 0.f, 0.f, 0.f, 0.f};
        #pragma unroll 4
        for (int k0 = 0; k0 < EXPERT_SIZE; k0 += 4) {
            v2f b, a0, a1;
            b.x  = vB[(size_t)(k0 + khalf + 0) * DMODEL + n0 + ncol];
            b.y  = vB[(size_t)(k0 + khalf + 1) * DMODEL + n0 + ncol];
            a0.x = Ss[mrow * SPITCH + k0 + khalf + 0];
            a0.y = Ss[mrow * SPITCH + k0 + khalf + 1];
            a1.x = Ss[(mrow + 16) * SPITCH + k0 + khalf + 0];
            a1.y = Ss[(mrow + 16) * SPITCH + k0 + khalf + 1];
            c0 = __builtin_amdgcn_wmma_f32_16x16x4_f32(
                     false, a0, false, b, (short)0, c0, false, false);
            c1 = __builtin_amdgcn_wmma_f32_16x16x4_f32(
                     false, a1, false, b, (short)0, c1, false, false);
        }
        // accumulate into out (each token gets exactly N_HEADS contributions)
        #pragma unroll
        for (int r = 0; r < 8; ++r) {
            int M = r + ((lane >> 4) << 3);
            int t0 = Ts[M];
            int t1 = Ts[M + 16];
            if (t0 >= 0)
                atomicAdd(&out[(size_t)t0 * DMODEL + n0 + ncol], c0[r]);
            if (t1 >= 0)
                atomicAdd(&out[(size_t)t1 * DMODEL + n0 + ncol], c1[r]);
        }
    }
}

// ---------------------------------------------------------------------------
// Launch
// ---------------------------------------------------------------------------
extern "C" void kernel_launch(void* const* d_in, const int* in_sizes, int n_in,
                              void* d_out, int out_size, void* d_ws, size_t ws_size,
                              hipStream_t stream) {
    const float* x      = (const float*)d_in[0];   // [T, d]
    const float* keys   = (const float*)d_in[1];   // [E, d, h]
    const float* values = (const float*)d_in[2];   // [E, h, d]
    const float* es     = (const float*)d_in[3];   // [E, d]
    // d_in[4] = n_heads (compile-time N_HEADS)

    float* out     = (float*)d_out;                // [T*d] + reg_loss
    float* out_reg = out + (size_t)TOKENS * DMODEL;

    // workspace layout (fits in ~1.1 MB)
    float* S       = (float*)d_ws;                        // [32]
    int*   cnt     = (int*)d_ws + 32;                     // [32]
    int*   toklist = (int*)d_ws + 64;                     // [32*CAP]
    float* wlist   = (float*)d_ws + 64 + N_EXPERTS * CAP; // [32*CAP]

    // 0) zero output + accumulators
    {
        int n = out_size;
        int blocks = (n + 255) / 256;
        moe_zero_kernel<<<blocks, 256, 0, stream>>>(out, n, S, cnt);
    }
    // 1) routing + dispatch-list build
    moe_routing_kernel<<<TOKENS / 8, 256, 0, stream>>>(x, es, S, cnt, toklist, wlist);
    // 2) regularizer
    moe_reg_finalize<<<1, 32, 0, stream>>>(S, out_reg);
    // 3) grouped expert GEMMs (WMMA, 32-token tiles)
    dim3 grid(N_EXPERTS, CAP / MTILE);
    moe_expert_gemm<<<grid, 512, 0, stream>>>(x, keys, values, cnt, toklist, wlist, out);
}